// GCN_46359876993041
// MI455X (gfx1250) — compile-verified
//
#include <hip/hip_runtime.h>

typedef __attribute__((ext_vector_type(2))) float v2f;
typedef __attribute__((ext_vector_type(8))) float v8f;

#define GCN_IN 128
#define GCN_HID 64
#define GCN_OUT 128

// ---------- degree / normalization ----------
__global__ void deg_init_kernel(float* __restrict__ deg, int n) {
  int i = blockIdx.x * blockDim.x + threadIdx.x;
  if (i < n) deg[i] = 1.0f;  // self-loop contributes 1 to every node's degree
}

__global__ void deg_edge_kernel(const int* __restrict__ dst, float* __restrict__ deg, int E) {
  int e = blockIdx.x * blockDim.x + threadIdx.x;
  if (e < E) atomicAdd(&deg[dst[e]], 1.0f);
}

__global__ void dinv_kernel(float* __restrict__ deg, int n) {
  int i = blockIdx.x * blockDim.x + threadIdx.x;
  if (i < n) deg[i] = rsqrtf(deg[i]);  // in place: deg -> d^-1/2
}

// ---------- dense GEMM via V_WMMA_F32_16X16X4_F32 with async-LDS weight staging ----------
// C[M,NN] = A[M,KK] * B[KK,NN]; M%16==0. One wave32 per 16x16 C tile, 8 waves/block.
// The whole weight matrix (KK*NN*4 = 32KB) is staged into LDS once per block using
// GLOBAL_LOAD_ASYNC_TO_LDS_B128 (ASYNCcnt), then all waves read B fragments from LDS.
template <int NN, int KK>
__global__ void wmma_gemm_f32_lds(const float* __restrict__ A, const float* __restrict__ B,
                                  float* __restrict__ C, int M) {
  __shared__ float sB[KK * NN];

  // cooperative async fill: 16B per transfer, coalesced, all threads participate
  {
    constexpr int nvec = (KK * NN) / 4;  // float4 transfers (2048)
    const float4* gsrc = (const float4*)B;
#pragma unroll
    for (int i = threadIdx.x; i < nvec; i += 256) {
      unsigned lds_addr = (unsigned)(size_t)(&sB[i * 4]);           // addr[31:0] = LDS offset
      unsigned long long gaddr = (unsigned long long)(size_t)(gsrc + i);
      asm volatile("global_load_async_to_lds_b128 %0, %1, off"
                   :
                   : "v"(lds_addr), "v"(gaddr)
                   : "memory");
    }
    asm volatile("s_wait_asynccnt 0" ::: "memory");
  }
  __syncthreads();

  const int lane = threadIdx.x & 31;
  const int wave_in_blk = threadIdx.x >> 5;
  const int tile = blockIdx.x * 8 + wave_in_blk;
  constexpr int ntiles = NN >> 4;
  const int total_tiles = (M >> 4) * ntiles;
  if (tile >= total_tiles) return;  // wave-uniform; EXEC all-1s past here; after the barrier

  const int tm = tile / ntiles;
  const int tn = tile - tm * ntiles;
  const int r16 = lane & 15;  // A: row in tile; B/C: column in tile
  const int kh = lane >> 4;   // K-half select per ISA 16x4 f32 A layout

  const float* Arow = A + (size_t)(tm * 16 + r16) * KK;
  const float* Blds = &sB[tn * 16 + r16];

  v8f acc = {0.f, 0.f, 0.f, 0.f, 0.f, 0.f, 0.f, 0.f};
#pragma unroll 4
  for (int k0 = 0; k0 < KK; k0 += 4) {
    const int ka = k0 + 2 * kh;
    v2f a, b;
    a.x = Arow[ka];
    a.y = Arow[ka + 1];
    b.x = Blds[ka * NN];        // ds_load from staged weights
    b.y = Blds[(ka + 1) * NN];
    acc = __builtin_amdgcn_wmma_f32_16x16x4_f32(
        /*neg_a=*/false, a, /*neg_b=*/false, b,
        /*c_mod=*/(short)0, acc, /*reuse_a=*/false, /*reuse_b=*/false);
  }

  // C/D layout: VGPR r holds M = 8*kh + r, N = r16
  float* Crow = C + (size_t)(tm * 16 + kh * 8) * NN + (size_t)(tn * 16 + r16);
#pragma unroll
  for (int r = 0; r < 8; ++r) Crow[(size_t)r * NN] = acc[r];
}

// ---------- aggregation ----------
// Seed accumulator with the self-loop term: agg[i,:] = dinv[i]^2 * h[i,:]
__global__ void agg_init_kernel(const float* __restrict__ h, const float* __restrict__ dinv,
                                float* __restrict__ agg, int n, int dim4) {
  int idx = blockIdx.x * blockDim.x + threadIdx.x;
  if (idx >= n * dim4) return;
  int i = idx / dim4;
  float s = dinv[i];
  s = s * s;
  float4 v = ((const float4*)h)[idx];
  float4 o;
  o.x = v.x * s; o.y = v.y * s; o.z = v.z * s; o.w = v.w * s;
  ((float4*)agg)[idx] = o;
}

// Scatter-add edge messages: agg[dst,:] += dinv[src]*dinv[dst]*h[src,:]
// One thread per (edge, 4-feature chunk); gather is a coalesced float4 row read,
// scatter is 4 f32 atomics that stay resident in the 192MB L2.
__global__ void edge_scatter_kernel(const int* __restrict__ src, const int* __restrict__ dst,
                                    const float* __restrict__ dinv, const float* __restrict__ h,
                                    float* __restrict__ agg, int E, int dim4) {
  int idx = blockIdx.x * blockDim.x + threadIdx.x;
  if (idx >= E * dim4) return;
  int e = idx / dim4;
  int c = idx - e * dim4;
  int s = src[e];
  int d = dst[e];
  float norm = dinv[s] * dinv[d];
  const int dim = dim4 << 2;
  float4 v = ((const float4*)(h + (size_t)s * dim))[c];
  float* ap = agg + (size_t)d * dim + (size_t)c * 4;
  atomicAdd(ap + 0, norm * v.x);
  atomicAdd(ap + 1, norm * v.y);
  atomicAdd(ap + 2, norm * v.z);
  atomicAdd(ap + 3, norm * v.w);
}

__global__ void bias_act_kernel(float* __restrict__ a, const float* __restrict__ bias,
                                int n, int dim, int relu) {
  int idx = blockIdx.x * blockDim.x + threadIdx.x;
  if (idx >= n * dim) return;
  int f = idx % dim;
  float v = a[idx] + bias[f];
  if (relu) v = fmaxf(v, 0.0f);
  a[idx] = v;
}

static inline int cdiv_i(int a, int b) { return (a + b - 1) / b; }

extern "C" void kernel_launch(void* const* d_in, const int* in_sizes, int n_in,
                              void* d_out, int out_size, void* d_ws, size_t ws_size,
                              hipStream_t stream) {
  const float* x  = (const float*)d_in[0];   // [N, 128]
  const int*   ei = (const int*)d_in[1];     // [2, E]
  const float* W1 = (const float*)d_in[2];   // [128, 64]
  const float* b1 = (const float*)d_in[3];   // [64]
  const float* W2 = (const float*)d_in[4];   // [64, 128]
  const float* b2 = (const float*)d_in[5];   // [128]
  float*       out = (float*)d_out;          // [N, 128]

  const int N = in_sizes[0] / GCN_IN;        // 50000
  const int E = in_sizes[1] / 2;             // 800000
  const int* src = ei;
  const int* dst = ei + E;

  // workspace carve-up (floats)
  float* dinv = (float*)d_ws;                // N (deg -> dinv in place)
  float* h1   = dinv + ((N + 255) & ~255);   // N*HID
  float* agg1 = h1 + (size_t)N * GCN_HID;    // N*HID
  float* h2   = agg1 + (size_t)N * GCN_HID;  // N*OUT

  const int B = 256;

  // normalization
  deg_init_kernel<<<cdiv_i(N, B), B, 0, stream>>>(dinv, N);
  deg_edge_kernel<<<cdiv_i(E, B), B, 0, stream>>>(dst, dinv, E);
  dinv_kernel<<<cdiv_i(N, B), B, 0, stream>>>(dinv, N);

  // layer 1: h1 = x @ W1   (M=50000, N=64, K=128) -> 12500 tiles, 8 waves/block
  {
    int tiles = (N >> 4) * (GCN_HID >> 4);
    wmma_gemm_f32_lds<GCN_HID, GCN_IN><<<cdiv_i(tiles, 8), 256, 0, stream>>>(x, W1, h1, N);
  }
  agg_init_kernel<<<cdiv_i(N * (GCN_HID / 4), B), B, 0, stream>>>(h1, dinv, agg1, N, GCN_HID / 4);
  edge_scatter_kernel<<<cdiv_i(E * (GCN_HID / 4), B), B, 0, stream>>>(src, dst, dinv, h1, agg1,
                                                                      E, GCN_HID / 4);
  bias_act_kernel<<<cdiv_i(N * GCN_HID, B), B, 0, stream>>>(agg1, b1, N, GCN_HID, 1);

  // layer 2: h2 = agg1 @ W2   (M=50000, N=128, K=64) -> 25000 tiles
  {
    int tiles = (N >> 4) * (GCN_OUT >> 4);
    wmma_gemm_f32_lds<GCN_OUT, GCN_HID><<<cdiv_i(tiles, 8), 256, 0, stream>>>(agg1, W2, h2, N);
  }
  agg_init_kernel<<<cdiv_i(N * (GCN_OUT / 4), B), B, 0, stream>>>(h2, dinv, out, N, GCN_OUT / 4);
  edge_scatter_kernel<<<cdiv_i(E * (GCN_OUT / 4), B), B, 0, stream>>>(src, dst, dinv, h2, out,
                                                                      E, GCN_OUT / 4);
  bias_act_kernel<<<cdiv_i(N * GCN_OUT, B), B, 0, stream>>>(out, b2, N, GCN_OUT, 0);
}